// ObjectVarianceLoss_8529805050142
// MI455X (gfx1250) — compile-verified
//
#include <hip/hip_runtime.h>
#include <hip/hip_bf16.h>

#define NSEG   32
#define NSTAT  5
#define MINPIX 50.0f

typedef float v2f __attribute__((ext_vector_type(2)));
typedef float v8f __attribute__((ext_vector_type(8)));

// ---------------------------------------------------------------------------
// Kernel 0: zero the (B*NSEG) x NSTAT f32 accumulator workspace.
// ---------------------------------------------------------------------------
__global__ void ovl_zero_ws(float* __restrict__ ws, int n) {
    int i = blockIdx.x * blockDim.x + threadIdx.x;
    if (i < n) ws[i] = 0.0f;
}

// ---------------------------------------------------------------------------
// Kernel 1: streaming segment statistics.
// Each block is bound to one image (bins are image-local: 32 segments).
// 8 wave-private LDS histograms (ds_add_f32 atomics, no inter-wave conflicts),
// flushed once per block via global_atomic_add_f32.
// float4/int4 loads (4 px/lane) + global_prefetch of the next tile.
// ---------------------------------------------------------------------------
__global__ __launch_bounds__(256) void ovl_seg_stats(
    const float* __restrict__ flow, const int* __restrict__ masks,
    float* __restrict__ ws, int HW, int BPI)
{
    __shared__ float hist[8][NSTAT][NSEG];   // 5120 B: wave-private histograms
    const int tid = threadIdx.x;
    const int w   = tid >> 5;                // wave32: wave id within block

    for (int t = tid; t < 8 * NSTAT * NSEG; t += 256)
        ((float*)hist)[t] = 0.0f;
    __syncthreads();

    const int img   = blockIdx.x / BPI;
    const int slice = blockIdx.x - img * BPI;

    const float4* __restrict__ u4p =
        reinterpret_cast<const float4*>(flow + (size_t)(2 * img)     * (size_t)HW);
    const float4* __restrict__ v4p =
        reinterpret_cast<const float4*>(flow + (size_t)(2 * img + 1) * (size_t)HW);
    const int4*   __restrict__ m4p =
        reinterpret_cast<const int4*>(masks + (size_t)img * (size_t)HW);

    const int nq     = HW >> 2;        // float4 elements per image plane
    const int stride = BPI * 256;      // float4 elements swept per grid-slice pass

    for (int q = slice * 256 + tid; q < nq; q += stride) {
        // Prefetch next tile into cache hierarchy (global_prefetch_b8).
        int qn = q + stride;
        if (qn < nq) {
            __builtin_prefetch((const void*)(u4p + qn), 0, 1);
            __builtin_prefetch((const void*)(v4p + qn), 0, 1);
            __builtin_prefetch((const void*)(m4p + qn), 0, 1);
        }

        float4 uu = u4p[q];
        float4 vv = v4p[q];
        int4   mm = m4p[q];

        float us[4] = {uu.x, uu.y, uu.z, uu.w};
        float vs[4] = {vv.x, vv.y, vv.z, vv.w};
        int   ms[4] = {mm.x, mm.y, mm.z, mm.w};

        #pragma unroll
        for (int k = 0; k < 4; ++k) {
            int   s = ms[k] & (NSEG - 1);
            float u = us[k];
            float v = vs[k];
            atomicAdd(&hist[w][0][s], 1.0f);
            atomicAdd(&hist[w][1][s], u);
            atomicAdd(&hist[w][2][s], u * u);
            atomicAdd(&hist[w][3][s], v);
            atomicAdd(&hist[w][4][s], v * v);
        }
    }
    __syncthreads();

    // Reduce the 8 wave-private copies and flush to global accumulators.
    for (int t = tid; t < NSTAT * NSEG; t += 256) {
        int stat = t >> 5;
        int seg  = t & (NSEG - 1);
        float sum = 0.0f;
        #pragma unroll
        for (int ww = 0; ww < 8; ++ww) sum += hist[ww][stat][seg];
        atomicAdd(&ws[(size_t)(img * NSEG + seg) * NSTAT + stat], sum);
    }
}

// ---------------------------------------------------------------------------
// Kernel 2: finalize. 512 threads = 512 (image,segment) bins.
// Per-bin unbiased variance; sum of valid variances via chained
// V_WMMA_F32_16X16X4_F32 (exact f32 math: D = A x Ones + D acts as a 64-way
// fused adder tree, accumulator carried across 8 WMMAs); valid count via
// ballot/popc.
// ---------------------------------------------------------------------------
__global__ __launch_bounds__(512) void ovl_finalize(
    const float* __restrict__ ws, float* __restrict__ out)
{
    __shared__ float vals[512];
    __shared__ int   cnts[16];

    const int tid  = threadIdx.x;
    const int lane = tid & 31;
    const int wid  = tid >> 5;

    const float* s = ws + (size_t)tid * NSTAT;
    float cnt = s[0], su = s[1], suu = s[2], sv = s[3], svv = s[4];

    float safe  = fmaxf(cnt, 1.0f);
    float denom = fmaxf(cnt - 1.0f, 1.0f);
    float var_u = (suu - su * su / safe) / denom;
    float var_v = (svv - sv * sv / safe) / denom;

    bool valid = (cnt >= MINPIX) && ((tid & (NSEG - 1)) != 0);
    vals[tid] = valid ? (var_u + var_v) : 0.0f;

    unsigned long long bm = __ballot(valid ? 1 : 0);
    if (lane == 0) cnts[wid] = __popcll(bm);
    __syncthreads();

    if (wid == 0) {
        // A-matrix f32 16x4 layout: lanes 0-15 hold row M=lane, K={0,1};
        // lanes 16-31 hold row M=lane-16, K={2,3}. B = all-ones 4x16, so
        // D[m][n] = row-sum of A (layout of B irrelevant when B == 1).
        const int half = lane >> 4;
        const int row  = lane & 15;
        v8f d = {};
        v2f ones = {1.0f, 1.0f};
        #pragma unroll
        for (int c = 0; c < 8; ++c) {
            int e = c * 64 + row * 4 + half * 2;
            v2f a = { vals[e], vals[e + 1] };
            d = __builtin_amdgcn_wmma_f32_16x16x4_f32(
                    /*neg_a=*/false, a, /*neg_b=*/false, ones,
                    /*c_mod=*/(short)0, d, /*reuse_a=*/false, /*reuse_b=*/false);
        }
        // Lane l holds D rows [8*half, 8*half+8) for column l%16; every column
        // is identical, so total = lane0 partial (rows 0-7) + lane16 partial.
        float partial = d[0] + d[1] + d[2] + d[3] + d[4] + d[5] + d[6] + d[7];
        float total = __shfl(partial, 0) + __shfl(partial, 16);
        if (lane == 0) {
            int n = 0;
            #pragma unroll
            for (int i = 0; i < 16; ++i) n += cnts[i];
            out[0] = (n > 0) ? (total / (float)n) : 0.0f;
        }
    }
}

// ---------------------------------------------------------------------------
extern "C" void kernel_launch(void* const* d_in, const int* in_sizes, int n_in,
                              void* d_out, int out_size, void* d_ws, size_t ws_size,
                              hipStream_t stream) {
    const float* flow  = (const float*)d_in[0];
    const int*   masks = (const int*)d_in[1];
    float*       out   = (float*)d_out;
    float*       ws    = (float*)d_ws;

    const int Bimg = 16;                      // reference: B=16
    const int HW   = in_sizes[1] / Bimg;      // 512*1024 pixels per image
    const int NWS  = Bimg * NSEG * NSTAT;     // 2560 f32 accumulators

    ovl_zero_ws<<<(NWS + 255) / 256, 256, 0, stream>>>(ws, NWS);

    const int BPI = 256;                      // blocks per image -> 4096 blocks
    ovl_seg_stats<<<dim3(Bimg * BPI), dim3(256), 0, stream>>>(flow, masks, ws, HW, BPI);

    ovl_finalize<<<dim3(1), dim3(512), 0, stream>>>(ws, out);
}